// MambaBlockTVM_36936718745993
// MI455X (gfx1250) — compile-verified
//
#include <hip/hip_runtime.h>
#include <hip/hip_bf16.h>

typedef float v2f __attribute__((ext_vector_type(2)));
typedef float v8f __attribute__((ext_vector_type(8)));

#define BATCH 4
#define SEQ   2048
#define DMOD  1024
#define NST   16
#define CHUNK 64
#define NCH   (SEQ / CHUNK)   // 32

// ws float layout:
//   [0 .. 5120)        AmatT[80][64]   (k-major: col k, row t) = [T | O] transposed
//   [5120 .. 6144)     Ut[64][16]      Ut[j][n] = (A^(63-j) B)[n]
//   [6144 .. 6400)     AcT[16][16]     AcT[k][m] = (A^64)[m][k]
//   [8192 ..)          S[B][NCH][D][16]
//   [8192+2M ..)       H[B][NCH][16][D]
#define WS_AMATT 0
#define WS_UT    (80 * 64)
#define WS_ACT   (80 * 64 + 64 * 16)
#define WS_S     8192
#define WS_H     (8192 + (size_t)BATCH * NCH * DMOD * NST)

// ---------------------------------------------------------------------------
// Kernel 0: precompute powers of A and the derived chunk matrices. One block.
// ---------------------------------------------------------------------------
__global__ __launch_bounds__(256) void k0_precompute(const float* __restrict__ A,
                                                     const float* __restrict__ Bvec,
                                                     float* __restrict__ ws) {
    __shared__ float cur[16][16];
    __shared__ float bb[16];
    __shared__ float kv[64];
    __shared__ float r[16];
    __shared__ float u[16];

    const int tid = threadIdx.x;           // 0..255
    const int m = tid >> 4, n = tid & 15;

    float* AmatT = ws + WS_AMATT;
    float* Ut    = ws + WS_UT;
    float* AcT   = ws + WS_ACT;

    if (tid < 16) bb[tid] = Bvec[tid];
    cur[m][n] = (m == n) ? 1.0f : 0.0f;    // A^0
    __syncthreads();

    for (int p = 0; p <= CHUNK; ++p) {
        // r = B^T A^p ; u = A^p B
        if (tid < 16) {
            float s = 0.f;
            for (int q = 0; q < 16; ++q) s += bb[q] * cur[q][tid];
            r[tid] = s;
        } else if (tid < 32) {
            const int mm = tid - 16;
            float s = 0.f;
            for (int q = 0; q < 16; ++q) s += cur[mm][q] * bb[q];
            u[mm] = s;
        }
        __syncthreads();

        if (p <= 63) {
            if (tid == 0) {                 // k_p = B^T A^p B
                float s = 0.f;
                for (int q = 0; q < 16; ++q) s += r[q] * bb[q];
                kv[p] = s;
            }
            if (tid < 16) Ut[(63 - p) * 16 + tid] = u[tid];   // A^p B at col j=63-p
        }
        if (p >= 1) {
            if (tid < 16) AmatT[(64 + tid) * 64 + (p - 1)] = r[tid]; // O row t=p-1
        }
        if (p == CHUNK) {
            AcT[n * 16 + m] = cur[m][n];   // AcT[k][m] = (A^64)[m][k]
        }

        // advance: cur <- A * cur
        float s = 0.f;
        for (int q = 0; q < 16; ++q) s += A[m * 16 + q] * cur[q][n];
        __syncthreads();
        cur[m][n] = s;
        __syncthreads();
    }
    __syncthreads();

    // Toeplitz part: AmatT[k][t] = (t >= k) ? kv[t-k] : 0
    for (int e = tid; e < 64 * 64; e += 256) {
        const int k = e >> 6, t = e & 63;
        AmatT[k * 64 + t] = (t >= k) ? kv[t - k] : 0.0f;
    }
}

// ---------------------------------------------------------------------------
// Kernel 1: S[b][c][d][n] = sum_j Ut[j][n] * x[b][c*64+j][d]   (WMMA GEMM)
//   per wave: one 16-channel tile; A-frag = X^T (M=d, K=j), B-frag = Ut.
// ---------------------------------------------------------------------------
__global__ __launch_bounds__(256) void k1_state_gemm(const float* __restrict__ x,
                                                     const float* __restrict__ Ut,
                                                     float* __restrict__ S) {
    const int lane = threadIdx.x & 31;
    const int wave = threadIdx.x >> 5;
    const int hi = lane >> 4;              // 0|1
    const int m = lane & 15;

    const int blk = blockIdx.x;            // B*NCH*8 blocks
    const int dslab = blk & 7;
    const int chunk = (blk >> 3) & (NCH - 1);
    const int b = blk >> 8;                // /(8*32)
    const int d0 = (dslab * 8 + wave) * 16;

    const float* xb = x + ((size_t)b * SEQ + (size_t)chunk * CHUNK) * DMOD;

    v8f acc = {};
#pragma unroll
    for (int k0 = 0; k0 < CHUNK; k0 += 4) {
        const int r0 = k0 + 2 * hi;
        v2f a, bf;
        a.x  = xb[(size_t)r0 * DMOD + d0 + m];
        a.y  = xb[(size_t)(r0 + 1) * DMOD + d0 + m];
        bf.x = Ut[r0 * 16 + m];
        bf.y = Ut[(r0 + 1) * 16 + m];
        acc = __builtin_amdgcn_wmma_f32_16x16x4_f32(false, a, false, bf,
                                                    (short)0, acc, false, false);
    }

    float* Sp = S + (((size_t)b * NCH + chunk) * DMOD + d0) * NST;
#pragma unroll
    for (int v = 0; v < 8; ++v)
        Sp[(v + 8 * hi) * NST + m] = acc[v];
}

// ---------------------------------------------------------------------------
// Kernel 2: chunk-level scan  H_{c+1} = H_c @ AcT + S_c  (16 channels / wave)
//   H (state at chunk start) stored coalesced as H[b][c][state][d] via LDS.
// ---------------------------------------------------------------------------
__global__ __launch_bounds__(256) void k2_scan(const float* __restrict__ AcT,
                                               const float* __restrict__ S,
                                               float* __restrict__ H) {
    __shared__ float hbuf[8][16][17];

    const int lane = threadIdx.x & 31;
    const int wave = threadIdx.x >> 5;
    const int hi = lane >> 4;
    const int m = lane & 15;

    const int gwave = blockIdx.x * 8 + wave;     // B*D/16 = 256 waves
    const int b = gwave >> 6;                    // /64
    const int d0 = (gwave & 63) * 16;

    // preload AcT fragments (B operand, K-major)
    v2f bf[4];
#pragma unroll
    for (int k0 = 0; k0 < 16; k0 += 4) {
        const int r0 = k0 + 2 * hi;
        bf[k0 >> 2].x = AcT[r0 * 16 + m];
        bf[k0 >> 2].y = AcT[(r0 + 1) * 16 + m];
    }

    v8f h = {};                                   // h[d=v+8*hi][state=m], zero init
    for (int c = 0; c < NCH; ++c) {
        // re-layout h (C/D layout) -> LDS as hbuf[d][state]
#pragma unroll
        for (int v = 0; v < 8; ++v)
            hbuf[wave][v + 8 * hi][m] = h[v];
        __syncthreads();

        // coalesced store of state at chunk start: H[b][c][state][d]
        float* Hp = H + (((size_t)b * NCH + c) * NST) * DMOD;
#pragma unroll
        for (int v = 0; v < 8; ++v)
            Hp[(size_t)(2 * v + hi) * DMOD + d0 + m] = hbuf[wave][m][2 * v + hi];

        // load S_c as accumulator
        const float* Sp = S + (((size_t)b * NCH + c) * DMOD + d0) * NST;
        v8f acc;
#pragma unroll
        for (int v = 0; v < 8; ++v)
            acc[v] = Sp[(v + 8 * hi) * NST + m];

#pragma unroll
        for (int k0 = 0; k0 < 16; k0 += 4) {
            const int r0 = k0 + 2 * hi;
            v2f a;
            a.x = hbuf[wave][m][r0];
            a.y = hbuf[wave][m][r0 + 1];
            acc = __builtin_amdgcn_wmma_f32_16x16x4_f32(false, a, false, bf[k0 >> 2],
                                                        (short)0, acc, false, false);
        }
        h = acc;
        __syncthreads();
    }
}

// ---------------------------------------------------------------------------
// Kernel 3: Y(64 x 1024) = [T | O](64 x 80) @ [X ; H](80 x 1024) per (b,chunk)
//   per wave: 4 t-tiles x 2 d-tiles (A-frag reuse); lower-triangular T ->
//   skip zero 16x16 K-blocks (t-tile `it` only needs K-blocks kb <= it).
// ---------------------------------------------------------------------------
__global__ __launch_bounds__(256) void k3_out_gemm(const float* __restrict__ x,
                                                   const float* __restrict__ AmatT,
                                                   const float* __restrict__ H,
                                                   float* __restrict__ y) {
    __shared__ float Asm[80 * 64];               // 20 KB, shared coefficients
    for (int e = threadIdx.x; e < 80 * 64; e += 256) Asm[e] = AmatT[e];
    __syncthreads();

    const int lane = threadIdx.x & 31;
    const int wave = threadIdx.x >> 5;
    const int hi = lane >> 4;
    const int m = lane & 15;

    const int blk = blockIdx.x;                  // B*NCH*4
    const int dquad = blk & 3;
    const int chunk = (blk >> 2) & (NCH - 1);
    const int b = blk >> 7;                      // /(4*32)
    const int d0 = (dquad * 8 + wave) * 32;      // two 16-wide d tiles: d0, d0+16
    const int d1 = d0 + 16;

    const float* xb = x + ((size_t)b * SEQ + (size_t)chunk * CHUNK) * DMOD;
    const float* Hp = H + (((size_t)b * NCH + chunk) * NST) * DMOD;

    v8f acc[2][4] = {{{}, {}, {}, {}}, {{}, {}, {}, {}}};

    // K = 0..63 : Toeplitz @ X, with triangular K-block skipping
#pragma unroll
    for (int kb = 0; kb < 4; ++kb) {
#pragma unroll
        for (int kk = 0; kk < 16; kk += 4) {
            const int r0 = kb * 16 + kk + 2 * hi;
            v2f b0, b1;
            b0.x = xb[(size_t)r0 * DMOD + d0 + m];
            b0.y = xb[(size_t)(r0 + 1) * DMOD + d0 + m];
            b1.x = xb[(size_t)r0 * DMOD + d1 + m];
            b1.y = xb[(size_t)(r0 + 1) * DMOD + d1 + m];
#pragma unroll
            for (int it = kb; it < 4; ++it) {
                v2f a;
                a.x = Asm[r0 * 64 + it * 16 + m];
                a.y = Asm[(r0 + 1) * 64 + it * 16 + m];
                acc[0][it] = __builtin_amdgcn_wmma_f32_16x16x4_f32(
                    false, a, false, b0, (short)0, acc[0][it], false, false);
                acc[1][it] = __builtin_amdgcn_wmma_f32_16x16x4_f32(
                    false, a, false, b1, (short)0, acc[1][it], false, false);
            }
        }
    }
    // K = 64..79 : O @ H (dense for all t-tiles)
#pragma unroll
    for (int k0 = 64; k0 < 80; k0 += 4) {
        const int r0 = k0 + 2 * hi;
        v2f b0, b1;
        b0.x = Hp[(size_t)(r0 - 64) * DMOD + d0 + m];
        b0.y = Hp[(size_t)(r0 - 63) * DMOD + d0 + m];
        b1.x = Hp[(size_t)(r0 - 64) * DMOD + d1 + m];
        b1.y = Hp[(size_t)(r0 - 63) * DMOD + d1 + m];
#pragma unroll
        for (int it = 0; it < 4; ++it) {
            v2f a;
            a.x = Asm[r0 * 64 + it * 16 + m];
            a.y = Asm[(r0 + 1) * 64 + it * 16 + m];
            acc[0][it] = __builtin_amdgcn_wmma_f32_16x16x4_f32(
                false, a, false, b0, (short)0, acc[0][it], false, false);
            acc[1][it] = __builtin_amdgcn_wmma_f32_16x16x4_f32(
                false, a, false, b1, (short)0, acc[1][it], false, false);
        }
    }

    float* yb = y + ((size_t)b * SEQ + (size_t)chunk * CHUNK) * DMOD;
#pragma unroll
    for (int it = 0; it < 4; ++it)
#pragma unroll
        for (int v = 0; v < 8; ++v) {
            const size_t row = (size_t)(it * 16 + v + 8 * hi) * DMOD;
            yb[row + d0 + m] = acc[0][it][v];
            yb[row + d1 + m] = acc[1][it][v];
        }
}

// ---------------------------------------------------------------------------
extern "C" void kernel_launch(void* const* d_in, const int* in_sizes, int n_in,
                              void* d_out, int out_size, void* d_ws, size_t ws_size,
                              hipStream_t stream) {
    (void)in_sizes; (void)n_in; (void)out_size; (void)ws_size;

    const float* x    = (const float*)d_in[0];
    const float* A    = (const float*)d_in[1];
    const float* Bvec = (const float*)d_in[2];
    float* y  = (float*)d_out;
    float* ws = (float*)d_ws;

    float* AmatT = ws + WS_AMATT;
    float* Ut    = ws + WS_UT;
    float* AcT   = ws + WS_ACT;
    float* S     = ws + WS_S;
    float* H     = ws + WS_H;

    k0_precompute<<<1, 256, 0, stream>>>(A, Bvec, ws);
    k1_state_gemm<<<BATCH * NCH * 8, 256, 0, stream>>>(x, Ut, S);
    k2_scan<<<(BATCH * (DMOD / 16)) / 8, 256, 0, stream>>>(AcT, S, H);
    k3_out_gemm<<<BATCH * NCH * 4, 256, 0, stream>>>(x, AmatT, H, y);
}